// Graph_protein_55353538511281
// MI455X (gfx1250) — compile-verified
//
#include <hip/hip_runtime.h>

typedef __attribute__((ext_vector_type(2))) float v2f;
typedef __attribute__((ext_vector_type(8))) float v8f;

#define WMMA_F32(a, b, c) \
    __builtin_amdgcn_wmma_f32_16x16x4_f32(false, (a), false, (b), (short)0, (c), false, false)

// ---------------- utility kernels ----------------

__global__ void fill0_kernel(float* __restrict__ p, long long n) {
    long long i = (long long)blockIdx.x * blockDim.x + threadIdx.x;
    long long stride = (long long)gridDim.x * blockDim.x;
    for (; i < n; i += stride) p[i] = 0.0f;
}

// pad columns: out[r][c] = (c < Fin) ? in[r][c] : 0 ; out is M x Fout
__global__ void pad_cols_kernel(const float* __restrict__ in, float* __restrict__ out,
                                int M, int Fin, int Fout) {
    long long i = (long long)blockIdx.x * blockDim.x + threadIdx.x;
    long long total = (long long)M * Fout;
    if (i >= total) return;
    int r = (int)(i / Fout);
    int c = (int)(i % Fout);
    out[i] = (c < Fin) ? in[(long long)r * Fin + c] : 0.0f;
}

// pad rows: out is Rout x C, rows >= Rin are zero
__global__ void pad_rows_kernel(const float* __restrict__ in, float* __restrict__ out,
                                int Rin, int Rout, int C) {
    long long i = (long long)blockIdx.x * blockDim.x + threadIdx.x;
    long long total = (long long)Rout * C;
    if (i >= total) return;
    int r = (int)(i / C);
    out[i] = (r < Rin) ? in[i] : 0.0f;
}

// pack K x Ncols row-major weight into WMMA B-fragment order:
// Bp[(k4*Ncols + n)*4 + r] = B[(k4*4 + r)*Ncols + n]
// so a lane's two B values (rows k+koff, k+koff+1 at col n) are contiguous.
__global__ void pack_b_kernel(const float* __restrict__ B, float* __restrict__ Bp,
                              int K, int Ncols) {
    long long i = (long long)blockIdx.x * blockDim.x + threadIdx.x;
    long long total = (long long)K * Ncols;
    if (i >= total) return;
    long long quad = i >> 2;                 // k4*Ncols + n
    int r = (int)(i & 3);
    int k4 = (int)(quad / Ncols);
    int n = (int)(quad - (long long)k4 * Ncols);
    Bp[i] = B[(long long)(k4 * 4 + r) * Ncols + n];
}

// ---------------- edge scatter-add (segment_sum over edges) ----------------
// one thread per (edge, 4-feature chunk); F must be a multiple of 4
__global__ void scatter_add_kernel(const float* __restrict__ feat,
                                   const long long* __restrict__ srcIdx,
                                   const long long* __restrict__ dstIdx,
                                   float* __restrict__ agg,
                                   long long E, int F) {
    int chunks = F >> 2;  // power-of-two (16/32/64) -> shifts
    long long tid = (long long)blockIdx.x * blockDim.x + threadIdx.x;
    long long total = E * chunks;
    if (tid >= total) return;
    long long e = tid / chunks;
    int c = (int)(tid - e * chunks);
    long long s = srcIdx[e];
    long long d = dstIdx[e];
    const float4 v = *(const float4*)(feat + s * F + (c << 2));
    float* dp = agg + d * F + (c << 2);
    atomicAdd(dp + 0, v.x);
    atomicAdd(dp + 1, v.y);
    atomicAdd(dp + 2, v.z);
    atomicAdd(dp + 3, v.w);
}

// ---------------- fused SAGE GEMM: D = relu(Aagg @ Bpl + bias + Ah @ Bpr) --
// Aagg, Ah, D: Mp x K / Mp x Ncols row-major, Mp % 32 == 0 (padded rows ok)
// Bpl, Bpr:    packed (pack_b_kernel) K x Ncols weights (K%8==0, Ncols%64==0)
// 4 waves per block; wave owns 32(M) x 64(N) of D: 2 M-subtiles x 4 N-subtiles
// = 8 accumulators. B fragments shared by both M-subtiles. Ping-pong double
// buffer (no cross-iteration copies): 32 WMMA + 24 b64 loads per 2 k-steps.

struct Frag {
    v2f a0g, a1g, a0h, a1h;          // A fragments: {subtile0,1} x {agg,h}
    v2f bl0, bl1, bl2, bl3;          // Bl fragments, 4 column sub-tiles
    v2f br0, br1, br2, br3;          // Br fragments
};

__device__ __forceinline__ Frag load_frags(const float* aP1, const float* aQ1,
                                           const float* aP2, const float* aQ2,
                                           const float* bPl, const float* bPr,
                                           int k, long long bstep) {
    Frag f;
    f.a0g = *(const v2f*)(aP1 + k);
    f.a1g = *(const v2f*)(aQ1 + k);
    f.a0h = *(const v2f*)(aP2 + k);
    f.a1h = *(const v2f*)(aQ2 + k);
    const float* p = bPl + (long long)(k >> 2) * bstep;
    f.bl0 = *(const v2f*)(p);
    f.bl1 = *(const v2f*)(p + 64);    // +16 cols * 4 packed floats
    f.bl2 = *(const v2f*)(p + 128);
    f.bl3 = *(const v2f*)(p + 192);
    const float* q = bPr + (long long)(k >> 2) * bstep;
    f.br0 = *(const v2f*)(q);
    f.br1 = *(const v2f*)(q + 64);
    f.br2 = *(const v2f*)(q + 128);
    f.br3 = *(const v2f*)(q + 192);
    return f;
}

__device__ __forceinline__ void wmma_block(const Frag& f,
        v8f& a00, v8f& a01, v8f& a02, v8f& a03,
        v8f& a10, v8f& a11, v8f& a12, v8f& a13) {
    a00 = WMMA_F32(f.a0g, f.bl0, a00);
    a01 = WMMA_F32(f.a0g, f.bl1, a01);
    a02 = WMMA_F32(f.a0g, f.bl2, a02);
    a03 = WMMA_F32(f.a0g, f.bl3, a03);
    a10 = WMMA_F32(f.a1g, f.bl0, a10);
    a11 = WMMA_F32(f.a1g, f.bl1, a11);
    a12 = WMMA_F32(f.a1g, f.bl2, a12);
    a13 = WMMA_F32(f.a1g, f.bl3, a13);
    a00 = WMMA_F32(f.a0h, f.br0, a00);
    a01 = WMMA_F32(f.a0h, f.br1, a01);
    a02 = WMMA_F32(f.a0h, f.br2, a02);
    a03 = WMMA_F32(f.a0h, f.br3, a03);
    a10 = WMMA_F32(f.a1h, f.br0, a10);
    a11 = WMMA_F32(f.a1h, f.br1, a11);
    a12 = WMMA_F32(f.a1h, f.br2, a12);
    a13 = WMMA_F32(f.a1h, f.br3, a13);
}

__global__ __launch_bounds__(128) void sage_gemm_kernel(
        const float* __restrict__ Aagg, const float* __restrict__ Bpl,
        const float* __restrict__ Ah,   const float* __restrict__ Bpr,
        const float* __restrict__ bias, float* __restrict__ D,
        int MtilePairs, int Ncols, int K) {
    const int lane = threadIdx.x & 31;
    const int wave = threadIdx.x >> 5;
    const int wslot = blockIdx.x * 4 + wave;   // wave-uniform
    if (wslot >= MtilePairs) return;           // whole wave exits together
    const int tm0 = wslot * 2;                 // two consecutive 16-row tiles
    const int n0 = blockIdx.y * 64;            // 4 column sub-tiles of 16

    const int l15 = lane & 15;
    const int koff = (lane >> 4) << 1;         // 0 for lanes 0-15, 2 for 16-31

    // A 16x4 fp32 fragment: lane<16 -> K={k,k+1}; lane>=16 -> K={k+2,k+3}
    const long long row0 = (long long)tm0 * 16 + l15;
    const float* aP1 = Aagg + row0 * K + koff;       // M-subtile 0, agg
    const float* aP2 = Ah   + row0 * K + koff;       // M-subtile 0, h
    const float* aQ1 = aP1 + 16LL * K;               // M-subtile 1, agg
    const float* aQ2 = aP2 + 16LL * K;               // M-subtile 1, h
    const int ncol = n0 + l15;
    const long long bstep = (long long)Ncols * 4;    // packed floats per k-quad
    const float* bPl = Bpl + (long long)ncol * 4 + koff;
    const float* bPr = Bpr + (long long)ncol * 4 + koff;

    v8f a00 = {}, a01 = {}, a02 = {}, a03 = {};      // M-subtile 0 accumulators
    v8f a10 = {}, a11 = {}, a12 = {}, a13 = {};      // M-subtile 1 accumulators

    // ping-pong: f0 covers k, f1 covers k+4; K is a multiple of 8
    Frag f0 = load_frags(aP1, aQ1, aP2, aQ2, bPl, bPr, 0, bstep);
    Frag f1 = load_frags(aP1, aQ1, aP2, aQ2, bPl, bPr, 4, bstep);
    for (int k = 8; k < K; k += 8) {
        wmma_block(f0, a00, a01, a02, a03, a10, a11, a12, a13);
        f0 = load_frags(aP1, aQ1, aP2, aQ2, bPl, bPr, k, bstep);
        wmma_block(f1, a00, a01, a02, a03, a10, a11, a12, a13);
        f1 = load_frags(aP1, aQ1, aP2, aQ2, bPl, bPr, k + 4, bstep);
    }
    wmma_block(f0, a00, a01, a02, a03, a10, a11, a12, a13);
    wmma_block(f1, a00, a01, a02, a03, a10, a11, a12, a13);

    // C/D layout: vgpr v -> row = 16*tile + v + 8*(lane>=16), col = ncol + 16*j
    v8f accs[2][4] = {{a00, a01, a02, a03}, {a10, a11, a12, a13}};
    const int rsub = (lane >> 4) << 3;
    #pragma unroll
    for (int t = 0; t < 2; ++t) {
        const int rbase = (tm0 + t) * 16 + rsub;
        #pragma unroll
        for (int j = 0; j < 4; ++j) {
            const int c = ncol + j * 16;
            const float bv = bias[c];
            #pragma unroll
            for (int v = 0; v < 8; ++v) {
                float val = accs[t][j][v] + bv;
                D[(long long)(rbase + v) * Ncols + c] = val > 0.0f ? val : 0.0f;
            }
        }
    }
}

// ---------------- global mean pool (batch is sorted by graph id) ----------
// blockDim.x == F; thread owns feature f = threadIdx.x over node chunk
// [blockIdx.x*CH, ...); run-length accumulate, flush atomics on graph change.
__global__ void pool_accum_kernel(const float* __restrict__ h,
                                  const long long* __restrict__ batch,
                                  float* __restrict__ sums,
                                  int N, int F, int CH) {
    const int f = threadIdx.x;
    int n = blockIdx.x * CH;
    const int n1 = (n + CH < N) ? n + CH : N;
    if (n >= N) return;
    int g = (int)batch[n];
    float acc = 0.0f;
    for (; n < n1; ++n) {
        int gn = (int)batch[n];
        if (gn != g) {
            atomicAdd(&sums[(long long)g * F + f], acc);
            acc = 0.0f;
            g = gn;
        }
        acc += h[(long long)n * F + f];
    }
    atomicAdd(&sums[(long long)g * F + f], acc);
}

__global__ void pool_count_kernel(const long long* __restrict__ batch,
                                  float* __restrict__ cnt, int N) {
    int n = blockIdx.x * blockDim.x + threadIdx.x;
    if (n < N) atomicAdd(&cnt[(int)batch[n]], 1.0f);
}

__global__ void pool_final_kernel(float* __restrict__ out, const float* __restrict__ cnt,
                                  int G, int F) {
    int tid = blockIdx.x * blockDim.x + threadIdx.x;
    if (tid >= G * F) return;
    float c = cnt[tid / F];
    out[tid] /= (c > 1.0f ? c : 1.0f);
}

// ---------------- launcher ----------------

static inline long long cdiv(long long a, long long b) { return (a + b - 1) / b; }

extern "C" void kernel_launch(void* const* d_in, const int* in_sizes, int n_in,
                              void* d_out, int out_size, void* d_ws, size_t ws_size,
                              hipStream_t stream) {
    const float*     x     = (const float*)d_in[0];
    const long long* ei    = (const long long*)d_in[1];
    const long long* batch = (const long long*)d_in[2];
    const float* W1l = (const float*)d_in[3];
    const float* b1  = (const float*)d_in[4];
    const float* W1r = (const float*)d_in[5];
    const float* W2l = (const float*)d_in[6];
    const float* b2  = (const float*)d_in[7];
    const float* W2r = (const float*)d_in[8];
    const float* W3l = (const float*)d_in[9];
    const float* b3  = (const float*)d_in[10];
    const float* W3r = (const float*)d_in[11];

    const int N = in_sizes[0] / 54;              // 50000
    const int Mp = (int)(cdiv(N, 32) * 32);      // padded to 32 rows (50016)
    const long long E = in_sizes[1] / 2;
    const long long* src = ei;
    const long long* dst = ei + E;
    const int G = 128;                           // NUM_GRAPHS
    const int pairs = Mp / 32;                   // 16-row tile pairs

    // workspace layout (floats); node-feature buffers padded to Mp rows.
    // Pad-row garbage never propagates: GEMM row r depends only on A row r,
    // edges index nodes < N, pool reads rows < N.
    float* ws    = (float*)d_ws;
    float* xp    = ws;  ws += (long long)Mp * 64;   // padded x
    float* w1lp  = ws;  ws += 64 * 128;             // row-padded W1l
    float* w1rp  = ws;  ws += 64 * 128;             // row-padded W1r
    float* w1lpk = ws;  ws += 64 * 128;             // packed weights
    float* w1rpk = ws;  ws += 64 * 128;
    float* w2lpk = ws;  ws += 128 * 256;
    float* w2rpk = ws;  ws += 128 * 256;
    float* w3lpk = ws;  ws += 256 * 256;
    float* w3rpk = ws;  ws += 256 * 256;
    float* agg   = ws;  ws += (long long)Mp * 256;  // reused per layer
    float* h1    = ws;  ws += (long long)Mp * 128;
    float* h2    = ws;  ws += (long long)Mp * 256;
    float* h3    = ws;  ws += (long long)Mp * 256;
    float* cnt   = ws;  ws += 256;

    const int TB = 256;

    // ---- prep: pad x (54->64 cols), pad W1 (54->64 rows), pack all weights
    pad_cols_kernel<<<(unsigned)cdiv((long long)N * 64, TB), TB, 0, stream>>>(x, xp, N, 54, 64);
    pad_rows_kernel<<<(unsigned)cdiv(64 * 128, TB), TB, 0, stream>>>(W1l, w1lp, 54, 64, 128);
    pad_rows_kernel<<<(unsigned)cdiv(64 * 128, TB), TB, 0, stream>>>(W1r, w1rp, 54, 64, 128);
    pack_b_kernel<<<(unsigned)cdiv(64 * 128, TB), TB, 0, stream>>>(w1lp, w1lpk, 64, 128);
    pack_b_kernel<<<(unsigned)cdiv(64 * 128, TB), TB, 0, stream>>>(w1rp, w1rpk, 64, 128);
    pack_b_kernel<<<(unsigned)cdiv(128 * 256, TB), TB, 0, stream>>>(W2l, w2lpk, 128, 256);
    pack_b_kernel<<<(unsigned)cdiv(128 * 256, TB), TB, 0, stream>>>(W2r, w2rpk, 128, 256);
    pack_b_kernel<<<(unsigned)cdiv(256 * 256, TB), TB, 0, stream>>>(W3l, w3lpk, 256, 256);
    pack_b_kernel<<<(unsigned)cdiv(256 * 256, TB), TB, 0, stream>>>(W3r, w3rpk, 256, 256);

    // ---- layer 1 ----
    fill0_kernel<<<2048, TB, 0, stream>>>(agg, (long long)N * 64);
    scatter_add_kernel<<<(unsigned)cdiv(E * 16, TB), TB, 0, stream>>>(xp, src, dst, agg, E, 64);
    {
        dim3 grid((unsigned)cdiv(pairs, 4), 128 / 64);
        sage_gemm_kernel<<<grid, 128, 0, stream>>>(agg, w1lpk, xp, w1rpk, b1, h1, pairs, 128, 64);
    }

    // ---- layer 2 ----
    fill0_kernel<<<2048, TB, 0, stream>>>(agg, (long long)N * 128);
    scatter_add_kernel<<<(unsigned)cdiv(E * 32, TB), TB, 0, stream>>>(h1, src, dst, agg, E, 128);
    {
        dim3 grid((unsigned)cdiv(pairs, 4), 256 / 64);
        sage_gemm_kernel<<<grid, 128, 0, stream>>>(agg, w2lpk, h1, w2rpk, b2, h2, pairs, 256, 128);
    }

    // ---- layer 3 ----
    fill0_kernel<<<2048, TB, 0, stream>>>(agg, (long long)N * 256);
    scatter_add_kernel<<<(unsigned)cdiv(E * 64, TB), TB, 0, stream>>>(h2, src, dst, agg, E, 256);
    {
        dim3 grid((unsigned)cdiv(pairs, 4), 256 / 64);
        sage_gemm_kernel<<<grid, 128, 0, stream>>>(agg, w3lpk, h2, w3rpk, b3, h3, pairs, 256, 256);
    }

    // ---- global mean pool ----
    fill0_kernel<<<64, TB, 0, stream>>>((float*)d_out, (long long)G * 256);
    fill0_kernel<<<1, TB, 0, stream>>>(cnt, G);
    pool_accum_kernel<<<(unsigned)cdiv(N, 32), 256, 0, stream>>>(h3, batch, (float*)d_out, N, 256, 32);
    pool_count_kernel<<<(unsigned)cdiv(N, TB), TB, 0, stream>>>(batch, cnt, N);
    pool_final_kernel<<<(unsigned)cdiv(G * 256, TB), TB, 0, stream>>>((float*)d_out, cnt, G, 256);
}